// Gate_1735166788251
// MI455X (gfx1250) — compile-verified
//
#include <hip/hip_runtime.h>

typedef __attribute__((ext_vector_type(2))) float v2f;
typedef __attribute__((ext_vector_type(4))) float v4f;
typedef __attribute__((ext_vector_type(8))) float v8f;
typedef int g4i __attribute__((vector_size(16)));   // matches builtin's vector_size(16) int

#define DIM      2048
#define NEXP     64
#define TOPK     6
#define NTOK     16384
#define KCHUNK   64
#define NCHUNK   (DIM / KCHUNK)      // 32
#define WPITCH   (KCHUNK + 4)        // 68 floats: bank = (4e + k) % 64, conflict-free
#define LPITCH   68
#define WAVES    4
#define BLOCK    (WAVES * 32)
#define TPW      16                  // tokens per wave

#if __has_builtin(__builtin_amdgcn_global_load_async_to_lds_b128)
#define USE_ASYNC 1
#else
#define USE_ASYNC 0
#endif

__device__ __forceinline__ void wait_async0() {
#if __has_builtin(__builtin_amdgcn_s_wait_asynccnt)
    __builtin_amdgcn_s_wait_asynccnt(0);
#else
    asm volatile("s_wait_asynccnt 0x0" ::: "memory");
#endif
}

__global__ __launch_bounds__(BLOCK) void moe_gate_wmma_async(
    const float* __restrict__ x,
    const float* __restrict__ w,
    float* __restrict__ wout,
    int*   __restrict__ iout)
{
    __shared__ float wbuf[2][NEXP * WPITCH];     // 34816 B, double-buffered W chunk
    __shared__ float lgs[WAVES][TPW * LPITCH];   // 17408 B, per-wave logits scratch

    const int tid    = threadIdx.x;
    const int lane   = tid & 31;
    const int wave   = tid >> 5;
    const int token0 = blockIdx.x * (WAVES * TPW) + wave * TPW;

    const int row  = lane & 15;           // A: token row / B: expert col
    const int koff = (lane >> 4) * 2;     // lane half -> K pair {0,1} vs {2,3}

    const float* xa = x + (size_t)(token0 + row) * DIM + koff;

    v8f c0 = {}, c1 = {}, c2 = {}, c3 = {};

    // Cooperative stage of W[:, k0:k0+KCHUNK] into wbuf[dbuf] (1024 x 16B pieces).
    auto stage = [&](int dbuf, int k0) {
        #pragma unroll
        for (int i = 0; i < (NEXP * KCHUNK / 4) / BLOCK; ++i) {   // 8 per thread
            int p   = tid + i * BLOCK;
            int e   = p >> 4;          // expert 0..63
            int cq  = p & 15;          // 16B piece within the 64-float row
            const float* src = w + (size_t)e * DIM + k0 + cq * 4;
            float*       dst = &wbuf[dbuf][e * WPITCH + cq * 4];
#if USE_ASYNC
            __builtin_amdgcn_global_load_async_to_lds_b128(
                (__attribute__((address_space(1))) g4i*)src,
                (__attribute__((address_space(3))) g4i*)dst, 0, 0);
#else
            *(v4f*)dst = *(const v4f*)src;
#endif
        }
    };

    stage(0, 0);

    for (int c = 0; c < NCHUNK; ++c) {
        const int cur = c & 1;
#if USE_ASYNC
        wait_async0();                 // my share of chunk c landed in LDS
#endif
        __syncthreads();               // everyone's share landed; prev compute done

        if (c + 1 < NCHUNK) stage(cur ^ 1, (c + 1) * KCHUNK);

        const float* abase = xa + c * KCHUNK;
        const float* bbase = &wbuf[cur][row * WPITCH + koff];
        __builtin_prefetch(abase + KCHUNK, 0, 0);   // global_prefetch next A chunk

        #pragma unroll 8
        for (int k2 = 0; k2 < KCHUNK; k2 += 4) {
            v2f a  = *(const v2f*)(abase + k2);
            v2f b0 = *(const v2f*)(bbase + k2);
            v2f b1 = *(const v2f*)(bbase + 16 * WPITCH + k2);
            v2f b2 = *(const v2f*)(bbase + 32 * WPITCH + k2);
            v2f b3 = *(const v2f*)(bbase + 48 * WPITCH + k2);
            c0 = __builtin_amdgcn_wmma_f32_16x16x4_f32(false, a, false, b0, (short)0, c0, false, false);
            c1 = __builtin_amdgcn_wmma_f32_16x16x4_f32(false, a, false, b1, (short)0, c1, false, false);
            c2 = __builtin_amdgcn_wmma_f32_16x16x4_f32(false, a, false, b2, (short)0, c2, false, false);
            c3 = __builtin_amdgcn_wmma_f32_16x16x4_f32(false, a, false, b3, (short)0, c3, false, false);
        }
    }

    // Spill 16x64 logits: VGPR r, lanes 0-15 -> M=r, lanes 16-31 -> M=r+8; N = lane&15.
    {
        float* lg0 = &lgs[wave][0];
        const int mhalf = (lane >> 4) * 8;
        const int ncol  = lane & 15;
        #pragma unroll
        for (int r = 0; r < 8; ++r) {
            lg0[(mhalf + r) * LPITCH +  0 + ncol] = c0[r];
            lg0[(mhalf + r) * LPITCH + 16 + ncol] = c1[r];
            lg0[(mhalf + r) * LPITCH + 32 + ncol] = c2[r];
            lg0[(mhalf + r) * LPITCH + 48 + ncol] = c3[r];
        }
    }
    __syncthreads();

    // Softmax + top-6: lanes 0-15 each own one token.
    if (lane < 16) {
        const float* lg = &lgs[wave][lane * LPITCH];
        const int tok = token0 + lane;

        float m = lg[0];
        #pragma unroll 8
        for (int e = 1; e < NEXP; ++e) m = fmaxf(m, lg[e]);

        float sum = 0.f;
        #pragma unroll 8
        for (int e = 0; e < NEXP; ++e) sum += __expf(lg[e] - m);
        const float inv = 1.0f / sum;   // ROUTE_SCALE == 1.0

        unsigned long long taken = 0ull;
        #pragma unroll
        for (int j = 0; j < TOPK; ++j) {
            float best = -3.402823466e+38f;
            int   bi   = 0;
            for (int e = 0; e < NEXP; ++e) {
                float v = lg[e];
                bool free = ((taken >> e) & 1ull) == 0ull;
                if (free && v > best) { best = v; bi = e; }   // '>' -> lowest index on ties
            }
            taken |= (1ull << bi);
            wout[tok * TOPK + j] = __expf(best - m) * inv;
            iout[tok * TOPK + j] = bi;
        }
    }
}

extern "C" void kernel_launch(void* const* d_in, const int* in_sizes, int n_in,
                              void* d_out, int out_size, void* d_ws, size_t ws_size,
                              hipStream_t stream) {
    (void)in_sizes; (void)n_in; (void)d_ws; (void)ws_size; (void)out_size;
    const float* x = (const float*)d_in[0];
    const float* w = (const float*)d_in[1];
    float* wout = (float*)d_out;                 // [16384, 6] fp32 weights
    int*   iout = (int*)(wout + NTOK * TOPK);    // [16384, 6] int32 indices (bit-stored)

    moe_gate_wmma_async<<<dim3(NTOK / (WAVES * TPW)), dim3(BLOCK), 0, stream>>>(x, w, wout, iout);
}